// GCN_27908697490049
// MI455X (gfx1250) — compile-verified
//
#include <hip/hip_runtime.h>
#include <hip/hip_bf16.h>

// ---------------------------------------------------------------------------
// GraphConv x3 + mean pool + linear for MI455X (gfx1250, wave32).
// Dense node transforms use V_WMMA_F32_16X16X4_F32 with a fused
// A=[agg|x] (16x16) x B=[rel_w;root_w] (16x16) K=16 chain (4 WMMAs/tile).
// ---------------------------------------------------------------------------

typedef __attribute__((ext_vector_type(2))) float v2f;
typedef __attribute__((ext_vector_type(8))) float v8f;

#define N_NODES   200000
#define N_EDGES   6400000
#define N_GRAPHS  1000
#define H_FEAT    8
#define N_CLASSES 3

// ---- utility kernels -------------------------------------------------------

__global__ void k_zero_f32(float* __restrict__ p, int n) {
    int i = blockIdx.x * blockDim.x + threadIdx.x;
    if (i < n) p[i] = 0.0f;
}

// x: [N,7] -> xp: [N,8] with column 7 zeroed (uniform stride for all layers)
__global__ void k_pad_x(const float* __restrict__ x, float* __restrict__ xp, int n8) {
    int i = blockIdx.x * blockDim.x + threadIdx.x;
    if (i >= n8) return;
    int node = i >> 3, f = i & 7;
    xp[i] = (f < 7) ? x[node * 7 + f] : 0.0f;
}

// Pack combined B = [rel_w (din x 8) ; root_w (din x 8)] into 16x16, zero-padded.
// Row-major Bp[k*16 + n].
__global__ void k_pack_b(const float* __restrict__ rel_w,
                         const float* __restrict__ root_w,
                         int din, float* __restrict__ Bp) {
    int i = threadIdx.x;          // one block of 256 threads
    int k = i >> 4, n = i & 15;
    float v = 0.0f;
    if (n < H_FEAT) {
        if (k < 8) { if (k < din)      v = rel_w [k * H_FEAT + n]; }
        else       { int kk = k - 8; if (kk < din) v = root_w[kk * H_FEAT + n]; }
    }
    Bp[i] = v;
}

// ---- edge scatter: agg[dst] += xin[src]  (rows are stride-8 f32) -----------

__global__ void k_scatter_add(const int* __restrict__ src,
                              const int* __restrict__ dst,
                              const float* __restrict__ xin,
                              float* __restrict__ agg, int nEdges) {
    int e = blockIdx.x * blockDim.x + threadIdx.x;
    if (e >= nEdges) return;
    int s = src[e];
    int d = dst[e];
    const float4* sp = (const float4*)(xin + (long)s * 8);   // global_load_b128 x2 (L2-resident)
    float4 v0 = sp[0];
    float4 v1 = sp[1];
    float* dp = agg + (long)d * 8;
    atomicAdd(dp + 0, v0.x); atomicAdd(dp + 1, v0.y);
    atomicAdd(dp + 2, v0.z); atomicAdd(dp + 3, v0.w);
    atomicAdd(dp + 4, v1.x); atomicAdd(dp + 5, v1.y);
    atomicAdd(dp + 6, v1.z); atomicAdd(dp + 7, v1.w);
}

// ---- fused node transform via WMMA f32 16x16x4 -----------------------------
// out[16x8 tile] = [agg | xin] (16x16) @ [rel_w ; root_w] (16x16) + bias, opt LeakyReLU
// A layout (16x4 f32): lanes 0-15 hold M=lane, K={0(v.x),1(v.y)}; lanes 16-31 K={2,3}.
// B layout (4x16 f32): lanes 0-15 hold N=lane, K={0,1}; lanes 16-31 K={2,3}.
// D layout: VGPR r, lanes 0-15 -> (M=r, N=lane); lanes 16-31 -> (M=r+8, N=lane-16).

__global__ void k_gconv_wmma(const float* __restrict__ agg,   // [N,8]
                             const float* __restrict__ xin,   // [N,8]
                             const float* __restrict__ Bp,    // [16,16] packed
                             const float* __restrict__ bias,  // [8]
                             float* __restrict__ out,         // [N,8]
                             int nTiles, int leaky) {
    int wave = (blockIdx.x * blockDim.x + threadIdx.x) >> 5;  // wave32
    int lane = threadIdx.x & 31;
    if (wave >= nTiles) return;                               // wave-uniform exit

    int col  = lane & 15;          // node-in-tile for A, output column for B/D
    int hi   = lane >> 4;          // half-wave selector
    int klo  = hi * 2;             // K offset within each 4-wide chunk

    long node = (long)wave * 16 + col;
    const float* arow = agg + node * 8;
    const float* xrow = xin + node * 8;

    v8f c = {0.f, 0.f, 0.f, 0.f, 0.f, 0.f, 0.f, 0.f};
    // K = 0..7  : agg features against rel_w rows
    #pragma unroll
    for (int ch = 0; ch < 2; ++ch) {
        int k0 = ch * 4 + klo;
        v2f a = { arow[k0], arow[k0 + 1] };
        v2f b = { Bp[k0 * 16 + col], Bp[(k0 + 1) * 16 + col] };
        c = __builtin_amdgcn_wmma_f32_16x16x4_f32(false, a, false, b,
                                                  (short)0, c, false, false);
    }
    // K = 8..15 : x features against root_w rows
    #pragma unroll
    for (int ch = 0; ch < 2; ++ch) {
        int k0 = ch * 4 + klo;
        v2f a = { xrow[k0], xrow[k0 + 1] };
        v2f b = { Bp[(8 + k0) * 16 + col], Bp[(9 + k0) * 16 + col] };
        c = __builtin_amdgcn_wmma_f32_16x16x4_f32(false, a, false, b,
                                                  (short)0, c, false, false);
    }

    // Only output columns 0..7 are live (lanes 0-7 and 16-23).
    if (col < H_FEAT) {
        float bv = bias[col];
        long base = (long)wave * 16 + hi * 8;
        #pragma unroll
        for (int r = 0; r < 8; ++r) {
            float v = c[r] + bv;
            if (leaky) v = (v >= 0.f) ? v : 0.01f * v;
            out[(base + r) * 8 + col] = v;
        }
    }
}

// ---- global mean pool + final linear ---------------------------------------

__global__ void k_pool_scatter(const float* __restrict__ h,
                               const int* __restrict__ batch,
                               float* __restrict__ sums,   // [G,8]
                               float* __restrict__ cnts,   // [G]
                               int n8) {
    int i = blockIdx.x * blockDim.x + threadIdx.x;
    if (i >= n8) return;
    int node = i >> 3, f = i & 7;
    int g = batch[node];
    atomicAdd(&sums[g * H_FEAT + f], h[i]);
    if (f == 0) atomicAdd(&cnts[g], 1.0f);
}

__global__ void k_final_linear(const float* __restrict__ sums,
                               const float* __restrict__ cnts,
                               const float* __restrict__ lin_w,  // [8,3]
                               const float* __restrict__ lin_b,  // [3]
                               float* __restrict__ out) {        // [G,3]
    int i = blockIdx.x * blockDim.x + threadIdx.x;
    if (i >= N_GRAPHS * N_CLASSES) return;
    int g = i / N_CLASSES, cidx = i % N_CLASSES;
    float cnt = cnts[g];
    cnt = cnt > 1.0f ? cnt : 1.0f;
    float acc = 0.0f;
    #pragma unroll
    for (int f = 0; f < H_FEAT; ++f)
        acc += (sums[g * H_FEAT + f] / cnt) * lin_w[f * N_CLASSES + cidx];
    out[i] = acc + lin_b[cidx];
}

// ---------------------------------------------------------------------------

extern "C" void kernel_launch(void* const* d_in, const int* in_sizes, int n_in,
                              void* d_out, int out_size, void* d_ws, size_t ws_size,
                              hipStream_t stream) {
    const float* x        = (const float*)d_in[0];
    const int*   ei       = (const int*)  d_in[1];
    const int*   batch    = (const int*)  d_in[2];
    const float* rel_w1   = (const float*)d_in[3];
    const float* rel_b1   = (const float*)d_in[4];
    const float* root_w1  = (const float*)d_in[5];
    const float* rel_w2   = (const float*)d_in[6];
    const float* rel_b2   = (const float*)d_in[7];
    const float* root_w2  = (const float*)d_in[8];
    const float* rel_w3   = (const float*)d_in[9];
    const float* rel_b3   = (const float*)d_in[10];
    const float* root_w3  = (const float*)d_in[11];
    const float* lin_w    = (const float*)d_in[12];
    const float* lin_b    = (const float*)d_in[13];

    const int* src = ei;
    const int* dst = ei + N_EDGES;

    // Workspace layout (floats)
    float* ws   = (float*)d_ws;
    const int N8 = N_NODES * 8;           // 1.6M
    float* xp   = ws;                     // [N,8]
    float* agg  = xp  + N8;               // [N,8]
    float* hA   = agg + N8;               // [N,8]
    float* hB   = hA  + N8;               // [N,8]
    float* B1   = hB  + N8;               // 256
    float* B2   = B1  + 256;
    float* B3   = B2  + 256;
    float* sums = B3  + 256;              // [G,8]
    float* cnts = sums + N_GRAPHS * H_FEAT; // [G]

    const int TPB = 256;
    const int gN8   = (N8 + TPB - 1) / TPB;               // 6250
    const int gEdge = (N_EDGES + TPB - 1) / TPB;          // 25000
    const int nTiles = N_NODES / 16;                      // 12500 (exact)
    const int gWmma = (nTiles * 32 + TPB - 1) / TPB;      // 8 waves/block

    // prep: pad x, pack weights, zero pool accumulators
    k_pad_x<<<gN8, TPB, 0, stream>>>(x, xp, N8);
    k_pack_b<<<1, 256, 0, stream>>>(rel_w1, root_w1, 7, B1);
    k_pack_b<<<1, 256, 0, stream>>>(rel_w2, root_w2, 8, B2);
    k_pack_b<<<1, 256, 0, stream>>>(rel_w3, root_w3, 8, B3);
    k_zero_f32<<<(N_GRAPHS * (H_FEAT + 1) + TPB - 1) / TPB, TPB, 0, stream>>>(
        sums, N_GRAPHS * (H_FEAT + 1));

    // layer 1: in = xp, out = hA, LeakyReLU
    k_zero_f32<<<gN8, TPB, 0, stream>>>(agg, N8);
    k_scatter_add<<<gEdge, TPB, 0, stream>>>(src, dst, xp, agg, N_EDGES);
    k_gconv_wmma<<<gWmma, TPB, 0, stream>>>(agg, xp, B1, rel_b1, hA, nTiles, 1);

    // layer 2: in = hA, out = hB, LeakyReLU
    k_zero_f32<<<gN8, TPB, 0, stream>>>(agg, N8);
    k_scatter_add<<<gEdge, TPB, 0, stream>>>(src, dst, hA, agg, N_EDGES);
    k_gconv_wmma<<<gWmma, TPB, 0, stream>>>(agg, hA, B2, rel_b2, hB, nTiles, 1);

    // layer 3: in = hB, out = hA, no activation
    k_zero_f32<<<gN8, TPB, 0, stream>>>(agg, N8);
    k_scatter_add<<<gEdge, TPB, 0, stream>>>(src, dst, hB, agg, N_EDGES);
    k_gconv_wmma<<<gWmma, TPB, 0, stream>>>(agg, hB, B3, rel_b3, hA, nTiles, 0);

    // mean pool + classifier head
    k_pool_scatter<<<gN8, TPB, 0, stream>>>(hA, batch, sums, cnts, N8);
    k_final_linear<<<(N_GRAPHS * N_CLASSES + TPB - 1) / TPB, TPB, 0, stream>>>(
        sums, cnts, lin_w, lin_b, (float*)d_out);
}